// TtFalconAttentionDecode_16578573763576
// MI455X (gfx1250) — compile-verified
//
#include <hip/hip_runtime.h>
#include <hip/hip_bf16.h>
#include <math.h>

// ---------------------------------------------------------------------------
// Falcon decode attention (B=32, HIDDEN=4544, 71 heads x 64, kv_len=2048)
// Memory-bound: ~200MB streamed per call vs ~4 GFLOP. Strategy:
//   * fp32 WMMA (V_WMMA_F32_16X16X4_F32) for both big GEMMs
//   * split-K (8-way) + atomic f32 accumulation for HBM-streaming parallelism
//   * VALU/DS attention (KV reads are the cost, K-dim is only 64)
// ---------------------------------------------------------------------------

#define HIDDEN   4544
#define N_HEADS  71
#define HEAD_DIM 64
#define QKV_N    ((N_HEADS + 2) * HEAD_DIM)   // 4672
#define BATCH    32
#define KV_MAX   2048
#define SPLITK   8
#define KCHUNK   (HIDDEN / SPLITK)            // 568, multiple of 4

typedef __attribute__((ext_vector_type(2))) float v2f;
typedef __attribute__((ext_vector_type(8))) float v8f;

// ---------------------------------------------------------------------------
// GEMM: Y(32xN) += X(32xK) * W(KxN), K split across blockIdx.y.
// One wave owns one 16-column tile; two 16x16 f32 accumulators cover M=32.
// A 16x4 frag: lanes 0-15 -> K=k,k+1 ; lanes 16-31 -> K=k+2,k+3 (b64 load).
// B 4x16 frag: symmetric (lane&15 = column, lane>>4 selects K pair).
// C/D 16x16:   vgpr r, lane -> M = r + (lane>>4)*8, N = lane&15.
// ---------------------------------------------------------------------------
__global__ void __launch_bounds__(128)
gemm32_wmma(const float* __restrict__ X, const float* __restrict__ W,
            float* __restrict__ Y, int N, int K, int kchunk)
{
    const int lane  = threadIdx.x & 31;
    const int wave  = threadIdx.x >> 5;
    const int ntile = blockIdx.x * 4 + wave;
    const int n0    = ntile * 16;
    const int col   = n0 + (lane & 15);
    const int krow  = (lane >> 4) << 1;          // 0 or 2
    const int k0    = blockIdx.y * kchunk;

    const float* xr0 = X + (size_t)(lane & 15) * K + k0 + krow;
    const float* xr1 = xr0 + (size_t)16 * K;
    const float* wp  = W + (size_t)(k0 + krow) * N + col;

    v8f c0 = {};
    v8f c1 = {};

    for (int k = 0; k < kchunk; k += 4) {
        v2f b;
        b.x = wp[0];
        b.y = wp[N];
        v2f a0 = *(const v2f*)(xr0 + k);   // 8B-aligned: k,k0 multiples of 4, krow even
        v2f a1 = *(const v2f*)(xr1 + k);
        c0 = __builtin_amdgcn_wmma_f32_16x16x4_f32(false, a0, false, b,
                                                   (short)0, c0, false, false);
        c1 = __builtin_amdgcn_wmma_f32_16x16x4_f32(false, a1, false, b,
                                                   (short)0, c1, false, false);
        wp += (size_t)4 * N;
    }

    const int rbase = (lane >> 4) * 8;
#pragma unroll
    for (int r = 0; r < 8; ++r) {
        unsafeAtomicAdd(&Y[(size_t)(rbase + r) * N + col],      c0[r]);
        unsafeAtomicAdd(&Y[(size_t)(16 + rbase + r) * N + col], c1[r]);
    }
}

// ---------------------------------------------------------------------------
// RoPE: rotate q (71 heads) and the single k head at pos = past_len.
// v is a passthrough. Caches are NOT mutated; position `past` is served from
// k_new / v_new inside the attention kernel.
// ---------------------------------------------------------------------------
__global__ void __launch_bounds__(256)
rope_kernel(const float* __restrict__ qkv, const int* __restrict__ past_p,
            float* __restrict__ q_out, float* __restrict__ k_new,
            float* __restrict__ v_new)
{
    const int b   = blockIdx.x;
    const float pos = (float)(*past_p);
    const float* row = qkv + (size_t)b * QKV_N;

    for (int e = threadIdx.x; e < N_HEADS * HEAD_DIM; e += blockDim.x) {
        const int d = e & 63;
        const int i = d & 31;
        const float inv_freq = powf(10000.0f, -(float)i / 32.0f);
        const float ang = pos * inv_freq;
        const float c = cosf(ang), s = sinf(ang);
        const int h = e >> 6;
        const float qv = row[h * HEAD_DIM + d];
        const float rh = (d < 32) ? -row[h * HEAD_DIM + d + 32]
                                  :  row[h * HEAD_DIM + d - 32];
        q_out[(size_t)b * N_HEADS * HEAD_DIM + e] = qv * c + rh * s;
    }

    for (int d = threadIdx.x; d < HEAD_DIM; d += blockDim.x) {
        const int i = d & 31;
        const float inv_freq = powf(10000.0f, -(float)i / 32.0f);
        const float ang = pos * inv_freq;
        const float c = cosf(ang), s = sinf(ang);
        const float kv = row[N_HEADS * HEAD_DIM + d];
        const float rh = (d < 32) ? -row[N_HEADS * HEAD_DIM + d + 32]
                                  :  row[N_HEADS * HEAD_DIM + d - 32];
        k_new[b * HEAD_DIM + d] = kv * c + rh * s;
        v_new[b * HEAD_DIM + d] = row[(N_HEADS + 1) * HEAD_DIM + d];
    }
}

// ---------------------------------------------------------------------------
// Attention: one block (256 thr = 8 waves) per (batch, head).
// Phase 1: wave-per-row q.k dot (lane holds 2 dims, shfl reduce) -> LDS.
// Phase 2: block softmax.  Phase 3: p.V with coalesced V rows (4 l-groups).
// ---------------------------------------------------------------------------
__global__ void __launch_bounds__(256)
attn_kernel(const float* __restrict__ q, const float* __restrict__ k_cache,
            const float* __restrict__ v_cache, const float* __restrict__ k_new,
            const float* __restrict__ v_new, const float* __restrict__ mask,
            const int* __restrict__ past_p, float* __restrict__ ctx)
{
    const int bh = blockIdx.x;
    const int b  = bh / N_HEADS;
    const int h  = bh % N_HEADS;
    const int past  = *past_p;
    const int kvlen = past + 1;

    __shared__ float s_scores[KV_MAX];
    __shared__ float s_q[HEAD_DIM];
    __shared__ float s_red[8];

    const int tid = threadIdx.x, lane = tid & 31, wave = tid >> 5;

    if (tid < HEAD_DIM)
        s_q[tid] = q[((size_t)b * N_HEADS + h) * HEAD_DIM + tid];
    __syncthreads();

    const float q0 = s_q[2 * lane], q1 = s_q[2 * lane + 1];

    for (int l = wave; l < kvlen; l += 8) {
        const float* krow = (l == past) ? (k_new + b * HEAD_DIM)
                                        : (k_cache + ((size_t)b * KV_MAX + l) * HEAD_DIM);
        const float2 kk = *(const float2*)(krow + 2 * lane);
        float part = q0 * kk.x + q1 * kk.y;
#pragma unroll
        for (int off = 16; off > 0; off >>= 1)
            part += __shfl_down(part, off, 32);
        if (lane == 0)
            s_scores[l] = part * 0.125f + mask[(size_t)b * KV_MAX + l];
    }
    __syncthreads();

    // ---- softmax: max ----
    float m = -INFINITY;
    for (int l = tid; l < kvlen; l += 256) m = fmaxf(m, s_scores[l]);
#pragma unroll
    for (int off = 16; off > 0; off >>= 1)
        m = fmaxf(m, __shfl_down(m, off, 32));
    if (lane == 0) s_red[wave] = m;
    __syncthreads();
    if (tid == 0) {
        float mm = s_red[0];
        for (int w = 1; w < 8; ++w) mm = fmaxf(mm, s_red[w]);
        s_red[0] = mm;
    }
    __syncthreads();
    m = s_red[0];
    __syncthreads();

    // ---- softmax: exp + sum ----
    float sum = 0.0f;
    for (int l = tid; l < kvlen; l += 256) {
        const float p = __expf(s_scores[l] - m);
        s_scores[l] = p;
        sum += p;
    }
#pragma unroll
    for (int off = 16; off > 0; off >>= 1)
        sum += __shfl_down(sum, off, 32);
    if (lane == 0) s_red[wave] = sum;
    __syncthreads();
    if (tid == 0) {
        float ss = 0.0f;
        for (int w = 0; w < 8; ++w) ss += s_red[w];
        s_red[0] = ss;
    }
    __syncthreads();
    const float inv = 1.0f / s_red[0];

    // ---- p . V ----
    const int d = tid & 63, g = tid >> 6;   // 4 l-groups x 64 dims
    float acc = 0.0f;
    for (int l = g; l < kvlen; l += 4) {
        const float* vrow = (l == past) ? (v_new + b * HEAD_DIM)
                                        : (v_cache + ((size_t)b * KV_MAX + l) * HEAD_DIM);
        acc += s_scores[l] * vrow[d];
    }
    __syncthreads();
    s_scores[tid] = acc;                    // reuse low 256 slots as scratch
    __syncthreads();
    if (g == 0) {
        const float r = s_scores[d] + s_scores[64 + d] +
                        s_scores[128 + d] + s_scores[192 + d];
        ctx[((size_t)b * N_HEADS + h) * HEAD_DIM + d] = r * inv;
    }
}

// ---------------------------------------------------------------------------
extern "C" void kernel_launch(void* const* d_in, const int* in_sizes, int n_in,
                              void* d_out, int out_size, void* d_ws, size_t ws_size,
                              hipStream_t stream)
{
    const float* hidden  = (const float*)d_in[0];  // 1x1x32x4544
    const float* mask    = (const float*)d_in[1];  // 32x1x1x2048
    const float* w_qkv   = (const float*)d_in[2];  // 4544x4672
    const float* w_dense = (const float*)d_in[3];  // 4544x4544
    const float* k_cache = (const float*)d_in[4];  // 32x1x2048x64
    const float* v_cache = (const float*)d_in[5];  // 32x1x2048x64
    const int*   past    = (const int*)d_in[6];    // scalar (2047)
    float* out = (float*)d_out;                    // 32x4544

    // workspace carve-up (16B-aligned sections)
    char* ws = (char*)d_ws;
    float* qkv   = (float*)ws; ws += (size_t)BATCH * QKV_N * sizeof(float);            // 598 KB
    float* q_rot = (float*)ws; ws += (size_t)BATCH * N_HEADS * HEAD_DIM * sizeof(float); // 582 KB
    float* k_new = (float*)ws; ws += (size_t)BATCH * HEAD_DIM * sizeof(float);
    float* v_new = (float*)ws; ws += (size_t)BATCH * HEAD_DIM * sizeof(float);
    float* ctx   = (float*)ws; ws += (size_t)BATCH * N_HEADS * HEAD_DIM * sizeof(float);

    // zero accumulation targets (graph-capturable)
    hipMemsetAsync(qkv, 0, (size_t)BATCH * QKV_N * sizeof(float), stream);
    hipMemsetAsync(out, 0, (size_t)out_size * sizeof(float), stream);

    // 1) QKV projection: N=4672 -> 292 col-tiles = 73 blocks x 4 waves; split-K 8
    gemm32_wmma<<<dim3(73, SPLITK), 128, 0, stream>>>(hidden, w_qkv, qkv,
                                                      QKV_N, HIDDEN, KCHUNK);
    // 2) RoPE
    rope_kernel<<<BATCH, 256, 0, stream>>>(qkv, past, q_rot, k_new, v_new);
    // 3) Attention (one block per batch*head)
    attn_kernel<<<BATCH * N_HEADS, 256, 0, stream>>>(q_rot, k_cache, v_cache,
                                                     k_new, v_new, mask, past, ctx);
    // 4) Dense projection: N=4544 -> 284 col-tiles = 71 blocks x 4 waves; split-K 8
    gemm32_wmma<<<dim3(71, SPLITK), 128, 0, stream>>>(ctx, w_dense, out,
                                                      HIDDEN, HIDDEN, KCHUNK);
}